// CRLoss_5660766896696
// MI455X (gfx1250) — compile-verified
//
#include <hip/hip_runtime.h>
#include <stdint.h>

// Problem constants (match reference).
#define NB        8192         // B
#define THREADS   256          // 8 waves (wave32)
#define CHUNK_R   2            // rows per async LDS chunk (64KB)
#define NCHUNK    16           // chunks per strip
#define TR        (CHUNK_R*NCHUNK)   // 32 rows per block
#define CPT       (NB/THREADS)       // 32 columns owned per thread
#define F4_PER    ((CHUNK_R*NB/4)/THREADS) // 16 float4 async loads per thread/chunk

// Order-preserving float<->uint key (exact, makes atomicMax on floats
// deterministic and correct for negatives / -inf).
__device__ __forceinline__ unsigned fenc(float f) {
  unsigned u = __float_as_uint(f);
  return (u & 0x80000000u) ? ~u : (u | 0x80000000u);
}
__device__ __forceinline__ float fdec(unsigned k) {
  return __uint_as_float((k & 0x80000000u) ? (k ^ 0x80000000u) : ~k);
}
#define ENC_NEG_INF 0x007FFFFFu  // fenc(-inf)

extern "C" __global__ void cr_init(unsigned* __restrict__ colmax) {
  int i = blockIdx.x * blockDim.x + threadIdx.x;
  if (i < NB) colmax[i] = ENC_NEG_INF;
}

// One pass over a 32-row strip: computes row contributions (image anchors)
// and per-column partial maxima (text anchors) from the SAME bytes.
// Data path: global -> LDS via gfx1250 async copies (non-temporal: the 256MB
// matrix is streamed exactly once, bigger than the 192MB L2 -> don't pollute),
// double buffered, synchronized with the split ASYNCcnt counter.
extern "C" __global__ void __launch_bounds__(THREADS)
cr_rowpass(const float* __restrict__ sim, const int* __restrict__ labels,
           const float* __restrict__ margin, unsigned* __restrict__ colmax,
           float* __restrict__ rowpart) {
  __shared__ __align__(16) float buf[2][CHUNK_R * NB];  // 2 x 64KB
  __shared__ float red[8];

  const int tid = threadIdx.x;
  const int lane = tid & 31;
  const int wid = tid >> 5;
  const int r0 = blockIdx.x * TR;
  const float* stripBase = sim + (size_t)r0 * NB;

  // Cache the labels of this thread's 32 owned columns (coalesced).
  int labc[CPT];
#pragma unroll
  for (int k = 0; k < CPT; ++k) labc[k] = labels[tid + THREADS * k];

  float colpart[CPT];
#pragma unroll
  for (int k = 0; k < CPT; ++k) colpart[k] = -INFINITY;

  const uint32_t ldsB0 = (uint32_t)(uintptr_t)&buf[0][0];  // low 32 bits of flat
  const uint32_t ldsB1 = (uint32_t)(uintptr_t)&buf[1][0];  // addr == LDS offset

  // Prologue: issue async fill of chunk 0 into buffer 0.
#pragma unroll
  for (int i = 0; i < F4_PER; ++i) {
    uint32_t l = ldsB0 + (uint32_t)(tid + THREADS * i) * 16u;
    const float* ga = stripBase + (size_t)(tid + THREADS * i) * 4;
    asm volatile("global_load_async_to_lds_b128 %0, %1, off th:TH_LOAD_NT"
                 :: "v"(l), "v"(ga) : "memory");
  }

  float blocksum = 0.f;  // meaningful on tid 0 only

  for (int chunk = 0; chunk < NCHUNK; ++chunk) {
    // Issue next chunk into the other buffer, then wait for current chunk.
    if (chunk + 1 < NCHUNK) {
      const float* g = stripBase + (size_t)(chunk + 1) * CHUNK_R * NB;
      const uint32_t lb = ((chunk + 1) & 1) ? ldsB1 : ldsB0;
#pragma unroll
      for (int i = 0; i < F4_PER; ++i) {
        uint32_t l = lb + (uint32_t)(tid + THREADS * i) * 16u;
        const float* ga = g + (size_t)(tid + THREADS * i) * 4;
        asm volatile("global_load_async_to_lds_b128 %0, %1, off th:TH_LOAD_NT"
                     :: "v"(l), "v"(ga) : "memory");
      }
      // Async loads complete in order: <=16 outstanding => current chunk done.
      asm volatile("s_wait_asynccnt 16" ::: "memory");
    } else {
      asm volatile("s_wait_asynccnt 0" ::: "memory");
    }
    __syncthreads();  // all waves' async fills visible

    const float* lbuf = (chunk & 1) ? &buf[1][0] : &buf[0][0];
    for (int rr = 0; rr < CHUNK_R; ++rr) {
      const int r = r0 + chunk * CHUNK_R + rr;
      const int labR = labels[r];
      float rmax = -INFINITY;
#pragma unroll
      for (int k = 0; k < CPT; ++k) {
        float v = lbuf[rr * NB + tid + THREADS * k];
        if (labc[k] != labR) {            // symmetric negative mask
          rmax = fmaxf(rmax, v);          // hardest negative for row r
          colpart[k] = fmaxf(colpart[k], v);  // partial hardest-neg for column
        }
      }
      // Block max-reduction of rmax (wave shfl, then 8 wave leaders).
#pragma unroll
      for (int off = 16; off > 0; off >>= 1)
        rmax = fmaxf(rmax, __shfl_xor(rmax, off, 32));
      if (lane == 0) red[wid] = rmax;
      __syncthreads();
      if (tid == 0) {
        float m = red[0];
#pragma unroll
        for (int w = 1; w < 8; ++w) m = fmaxf(m, red[w]);
        float ap = lbuf[rr * NB + r];     // diagonal element lives in this row
        float mg = margin[r];
        if (m > ap) blocksum += mg + m - ap;  // relu is identity when valid
      }
      __syncthreads();  // red[] reuse + buffer-reuse safety before next issue
    }
  }

  if (tid == 0) rowpart[blockIdx.x] = blocksum;

  // Merge per-column partial maxima globally (exact integer max => deterministic).
#pragma unroll
  for (int k = 0; k < CPT; ++k)
    atomicMax(&colmax[tid + THREADS * k], fenc(colpart[k]));
}

// Finalize: column (text-anchor) contributions + row partial sums, fixed order.
extern "C" __global__ void __launch_bounds__(THREADS)
cr_final(const float* __restrict__ sim, const float* __restrict__ margin,
         const unsigned* __restrict__ colmax, const float* __restrict__ rowpart,
         int nblocks, float* __restrict__ out) {
  __shared__ float red[8];
  const int tid = threadIdx.x;
  float s = 0.f;
  for (int c = tid; c < NB; c += THREADS) {
    float an = fdec(colmax[c]);
    float ap = sim[(size_t)c * NB + c];
    float mg = margin[c];
    if (an > ap) s += mg + an - ap;
  }
  for (int i = tid; i < nblocks; i += THREADS) s += rowpart[i];
#pragma unroll
  for (int off = 16; off > 0; off >>= 1) s += __shfl_xor(s, off, 32);
  const int lane = tid & 31, wid = tid >> 5;
  if (lane == 0) red[wid] = s;
  __syncthreads();
  if (tid == 0) {
    float t = red[0];
#pragma unroll
    for (int w = 1; w < 8; ++w) t += red[w];
    out[0] = t;
  }
}

extern "C" void kernel_launch(void* const* d_in, const int* in_sizes, int n_in,
                              void* d_out, int out_size, void* d_ws, size_t ws_size,
                              hipStream_t stream) {
  const float* sim    = (const float*)d_in[0];
  const int*   labels = (const int*)d_in[1];   // harness delivers ints as int32
  const float* margin = (const float*)d_in[2];
  // d_in[3] is `semi` (== 0, hard-negative path) — ignored.

  unsigned* colmax  = (unsigned*)d_ws;                          // 8192 keys
  float*    rowpart = (float*)((char*)d_ws + NB * sizeof(unsigned)); // 256 sums
  const int nblocks = NB / TR;  // 256 strips

  cr_init<<<(NB + 255) / 256, 256, 0, stream>>>(colmax);
  cr_rowpass<<<nblocks, THREADS, 0, stream>>>(sim, labels, margin, colmax, rowpart);
  cr_final<<<1, THREADS, 0, stream>>>(sim, margin, colmax, rowpart, nblocks,
                                      (float*)d_out);
}